// DynamicKVCache_45861660787371
// MI455X (gfx1250) — compile-verified
//
#include <hip/hip_runtime.h>

// ---------------------------------------------------------------------------
// DynamicKVCache.update for B=8, L=4096, T=1, H=32, D=128 (f32).
// Pure streaming copy: ~2.15 GB of traffic, HBM-bound (~92 us @ 23.3 TB/s).
// Bulk path: CDNA5 async global<->LDS DMA (ASYNCcnt-tracked b128 transfers),
// fallback: non-temporal b128 VGPR copy. Both issue 16 B/lane accesses.
// ---------------------------------------------------------------------------

typedef __attribute__((ext_vector_type(4))) float v4f; // 16-byte data vector
typedef __attribute__((ext_vector_type(4))) int   v4i; // builtin's b128 type

static constexpr int  BATCH        = 8;
static constexpr long L_CACHE      = 4096;               // cached seq len
static constexpr int  HD4          = (32 * 128) / 4;     // 1024 float4 per (b,l) row
static constexpr long CACHE4_PER_B = L_CACHE * HD4;      // 4,194,304 float4
static constexpr long OUT4_PER_B   = (L_CACHE + 1) * HD4;// 4,195,328 float4
static constexpr int  THREADS      = 256;                // 8 wave32
static constexpr int  CHUNK        = 8;                  // float4 per thread

#define AS1 __attribute__((address_space(1)))
#define AS3 __attribute__((address_space(3)))

#if defined(__gfx1250__) &&                                                    \
    __has_builtin(__builtin_amdgcn_global_load_async_to_lds_b128) &&           \
    __has_builtin(__builtin_amdgcn_global_store_async_from_lds_b128) &&        \
    __has_builtin(__builtin_amdgcn_s_wait_asynccnt)
#define USE_ASYNC_LDS 1
#else
#define USE_ASYNC_LDS 0
#endif

// Bulk copy of the existing caches into their slots in the output.
// grid = (CACHE4_PER_B / (THREADS*CHUNK), BATCH, 2)  z: 0 = K, 1 = V
__global__ __launch_bounds__(THREADS) void kv_bulk_copy(
    const v4f* __restrict__ cache_k, const v4f* __restrict__ cache_v,
    v4f* __restrict__ out_k, v4f* __restrict__ out_v) {
  const int b = blockIdx.y;
  const v4f* __restrict__ src =
      (blockIdx.z == 0 ? cache_k : cache_v) + (long)b * CACHE4_PER_B;
  v4f* __restrict__ dst =
      (blockIdx.z == 0 ? out_k : out_v) + (long)b * OUT4_PER_B;

  const long base = (long)blockIdx.x * (THREADS * CHUNK);

#if USE_ASYNC_LDS
  // CDNA5 async DMA: global -> LDS -> global, no VGPR landing, ASYNCcnt-tracked.
  __shared__ v4f stage[THREADS * CHUNK]; // 32 KiB staging (320 KB/WGP available)
#pragma unroll
  for (int u = 0; u < CHUNK; ++u) {
    const int s = u * THREADS + threadIdx.x; // lanes contiguous -> coalesced
    __builtin_amdgcn_global_load_async_to_lds_b128(
        (AS1 v4i*)(src + base + s), (AS3 v4i*)&stage[s], 0, 0);
  }
  __builtin_amdgcn_s_wait_asynccnt(0); // per-wave; lanes own disjoint LDS
#pragma unroll
  for (int u = 0; u < CHUNK; ++u) {
    const int s = u * THREADS + threadIdx.x;
    __builtin_amdgcn_global_store_async_from_lds_b128(
        (AS1 v4i*)(dst + base + s), (AS3 v4i*)&stage[s], 0, 0);
  }
  // Trailing async stores are covered by S_ENDPGM's implicit wait-idle.
#else
  // Fallback: non-temporal b128 copy through VGPRs (th:NT both sides —
  // 1 GiB stream cannot be reused out of the 192 MB L2).
  v4f tmp[CHUNK];
#pragma unroll
  for (int u = 0; u < CHUNK; ++u)
    tmp[u] = __builtin_nontemporal_load(src + base + u * THREADS + threadIdx.x);
#pragma unroll
  for (int u = 0; u < CHUNK; ++u)
    __builtin_nontemporal_store(tmp[u], dst + base + u * THREADS + threadIdx.x);
#endif
}

// Append the T=1 new key/value row at l = L_CACHE.
// grid = (HD4 / THREADS, BATCH, 2)  z: 0 = K, 1 = V
__global__ __launch_bounds__(THREADS) void kv_append_row(
    const v4f* __restrict__ key, const v4f* __restrict__ value,
    v4f* __restrict__ out_k, v4f* __restrict__ out_v) {
  const int b = blockIdx.y;
  const int i = blockIdx.x * THREADS + threadIdx.x; // [0, HD4)
  const v4f* __restrict__ src = (blockIdx.z == 0 ? key : value) + (long)b * HD4;
  v4f* __restrict__ dst =
      (blockIdx.z == 0 ? out_k : out_v) + (long)b * OUT4_PER_B + CACHE4_PER_B;
  __builtin_nontemporal_store(__builtin_nontemporal_load(src + i), dst + i);
}

extern "C" void kernel_launch(void* const* d_in, const int* in_sizes, int n_in,
                              void* d_out, int out_size, void* d_ws,
                              size_t ws_size, hipStream_t stream) {
  (void)in_sizes; (void)n_in; (void)out_size; (void)d_ws; (void)ws_size;

  const v4f* cache_k = (const v4f*)d_in[0];
  const v4f* cache_v = (const v4f*)d_in[1];
  const v4f* key     = (const v4f*)d_in[2];
  const v4f* value   = (const v4f*)d_in[3];

  v4f* out_k = (v4f*)d_out;
  v4f* out_v = out_k + (size_t)BATCH * OUT4_PER_B; // new_v follows new_k flat

  dim3 grid_bulk((unsigned)(CACHE4_PER_B / (THREADS * CHUNK)), BATCH, 2);
  kv_bulk_copy<<<grid_bulk, THREADS, 0, stream>>>(cache_k, cache_v, out_k, out_v);

  dim3 grid_app((unsigned)(HD4 / THREADS), BATCH, 2);
  kv_append_row<<<grid_app, THREADS, 0, stream>>>(key, value, out_k, out_v);
}